// RobustFill_60713657696724
// MI455X (gfx1250) — compile-verified
//
#include <hip/hip_runtime.h>
#include <math.h>

typedef __attribute__((ext_vector_type(2))) float v2f;
typedef __attribute__((ext_vector_type(8))) float v8f;

namespace {
constexpr int kJ   = 4;
constexpr int kB   = 8192;
constexpr int kNB  = kJ * kB;    // 32768 fused batch
constexpr int kT   = 10;         // L_IN == L_OUT
constexpr int kV   = 11;
constexpr int kH   = 100;
constexpr int kG3  = 3 * kH;     // 300
constexpr int kEMB = 100;
}

__global__ void rf_zero(float* __restrict__ p, int n) {
  int i = blockIdx.x * blockDim.x + threadIdx.x;
  if (i < n) p[i] = 0.0f;
}

// Generic GEMM on the matrix pipes:
//   out[n][c] = sum_k X[n*lda + k] * W[c*K + k]   (+ bias[c], optional tanh)
// One wave per 16x16 C tile; V_WMMA_F32_16X16X4_F32, K stepped by 4.
// A frag (16x4 f32): lanes 0-15 hold K=k0,k0+1; lanes 16-31 hold K=k0+2,k0+3 (M = lane&15).
// B frag (4x16 f32): mirrored halves, N = lane&15.
// Padded B columns are CLAMPED (not predicated): column c of B only feeds column c
// of C, and padded C columns are never stored, so clamping is safe and keeps the
// K loop free of EXEC manipulation (WMMA requires EXEC all-1s anyway).
__global__ void rf_gemm_wmma(const float* __restrict__ X, int lda,
                             const float* __restrict__ W, int K,
                             int outCols, int rows,
                             float* __restrict__ out,
                             const float* __restrict__ bias, int doTanh) {
  int wave   = blockIdx.x * (blockDim.x >> 5) + (threadIdx.x >> 5);
  int tilesN = (outCols + 15) >> 4;
  int tilesM = rows >> 4;
  int tn = wave % tilesN;
  int tm = wave / tilesN;
  if (tm >= tilesM) return;            // whole-wave uniform: EXEC stays all-1s for WMMA

  int lane = threadIdx.x & 31;
  int half = lane >> 4;
  int l16  = lane & 15;
  int row  = tm * 16 + l16;            // A-matrix row for this lane
  int col  = tn * 16 + l16;            // B/C column for this lane
  bool colOK = col < outCols;
  int ccol = colOK ? col : (outCols - 1);   // clamp: keeps loads in-bounds & aligned

  const float* xrow = X + (size_t)row * lda;
  const float* wcol = W + (size_t)ccol * K;

  v8f acc = {0.f, 0.f, 0.f, 0.f, 0.f, 0.f, 0.f, 0.f};
  for (int k0 = 0; k0 < K; k0 += 4) {
    int kb = k0 + half * 2;
    v2f a = *(const v2f*)(xrow + kb);       // 8B aligned: lda,kb even
    v2f b = *(const v2f*)(wcol + kb);       // 8B aligned: ccol*K even
    acc = __builtin_amdgcn_wmma_f32_16x16x4_f32(false, a, false, b,
                                                (short)0, acc, false, false);
  }

  if (colOK) {
    float bs = bias ? bias[col] : 0.0f;
    int rbase = tm * 16 + half * 8;          // C layout: vgpr v -> M = v (+8 for hi half)
#pragma unroll
    for (int v = 0; v < 8; ++v) {
      float val = acc[v] + bs;
      if (doTanh) val = tanhf(val);
      out[(size_t)(rbase + v) * outCols + col] = val;
    }
  }
}

// PyTorch-GRUCell gate math. gi comes from a one-hot input -> row gather of Wih.
// idx_mode: 0 = SOS (idx = V-1), 1 = encoder (idxp = inputs + t*B, idx = idxp[j*T*B + b]),
//           2 = decoder (idxp = target + t*B, idx = idxp[b]).
__global__ void rf_gru_gate(const float* __restrict__ gh,
                            const float* __restrict__ hprev, int hstride,
                            float* __restrict__ hout, int ostride,
                            float* __restrict__ hbuf,
                            const float* __restrict__ Wih,
                            const float* __restrict__ bih,
                            const float* __restrict__ bhh,
                            const int* __restrict__ idxp, int idx_mode) {
  int tid = blockIdx.x * blockDim.x + threadIdx.x;
  if (tid >= kNB * kH) return;
  int n = tid / kH;
  int j = tid - n * kH;

  int idx;
  if (idx_mode == 0) {
    idx = kV - 1;
  } else if (idx_mode == 1) {
    int jj = n >> 13;            // n / B
    int b  = n & (kB - 1);
    idx = idxp[jj * (kT * kB) + b];
  } else {
    idx = idxp[n & (kB - 1)];
  }

  float gir = Wih[(size_t)j * kV + idx]            + bih[j];
  float giz = Wih[(size_t)(j + kH) * kV + idx]     + bih[j + kH];
  float gin = Wih[(size_t)(j + 2 * kH) * kV + idx] + bih[j + 2 * kH];

  const float* ghn = gh + (size_t)n * kG3;
  float ghr = ghn[j]          + bhh[j];
  float ghz = ghn[j + kH]     + bhh[j + kH];
  float ghh = ghn[j + 2 * kH] + bhh[j + 2 * kH];

  float r  = 1.0f / (1.0f + expf(-(gir + ghr)));
  float z  = 1.0f / (1.0f + expf(-(giz + ghz)));
  float nn = tanhf(gin + r * ghh);
  float hp = hprev[(size_t)n * hstride + j];
  float hv = (1.0f - z) * nn + z * hp;

  hout[(size_t)n * ostride + j] = hv;
  if (hbuf) hbuf[(size_t)n * kH + j] = hv;
}

// scores[t] = H[t,n,:].AP[n,:]; logs = log_softmax_t; c = sum_t logs[t]*H[t,n,:]
// (faithful to the reference quirk: log_softmax, not softmax, weights H).
// Writes p_aug[n] = [P[n] | c[n]].
__global__ void rf_attend(const float* __restrict__ Hbuf,
                          const float* __restrict__ AP,
                          const float* __restrict__ P,
                          float* __restrict__ paug) {
  int n = blockIdx.x * blockDim.x + threadIdx.x;
  if (n >= kNB) return;
  const float* ap = AP + (size_t)n * kH;

  float sc[kT];
#pragma unroll
  for (int t = 0; t < kT; ++t) {
    const float* hr = Hbuf + ((size_t)t * kNB + n) * kH;
    float acc = 0.0f;
    for (int e = 0; e < kH; ++e) acc += hr[e] * ap[e];
    sc[t] = acc;
  }
  float mx = sc[0];
#pragma unroll
  for (int t = 1; t < kT; ++t) mx = fmaxf(mx, sc[t]);
  float s = 0.0f;
#pragma unroll
  for (int t = 0; t < kT; ++t) s += expf(sc[t] - mx);
  float lse = mx + logf(s);
#pragma unroll
  for (int t = 0; t < kT; ++t) sc[t] -= lse;

  float* pa = paug + (size_t)n * (2 * kH);
  const float* pn = P + (size_t)n * kH;
  for (int e = 0; e < kH; ++e) {
    float c = 0.0f;
#pragma unroll
    for (int t = 0; t < kT; ++t) c += sc[t] * Hbuf[((size_t)t * kNB + n) * kH + e];
    pa[e]      = pn[e];
    pa[kH + e] = c;
  }
}

// m = max_j fc[j*B+b]; logits = m @ V_w^T + V_b; score[b] += log_softmax(logits)[tgt[b]]
__global__ void rf_out_step(const float* __restrict__ fc,
                            const float* __restrict__ Vw,
                            const float* __restrict__ Vb,
                            const int* __restrict__ tgt,
                            float* __restrict__ score) {
  __shared__ float mbuf[128 * kEMB];          // 51.2 KB
  int tid = threadIdx.x;
  int b   = blockIdx.x * 128 + tid;
  float* m = mbuf + tid * kEMB;

  for (int e = 0; e < kEMB; ++e) {
    float mv = -1e30f;
#pragma unroll
    for (int j = 0; j < kJ; ++j)
      mv = fmaxf(mv, fc[((size_t)j * kB + b) * kEMB + e]);
    m[e] = mv;
  }

  float logits[kV];
#pragma unroll
  for (int v = 0; v < kV; ++v) {
    float acc = Vb[v];
    for (int e = 0; e < kEMB; ++e) acc += Vw[(size_t)v * kEMB + e] * m[e];
    logits[v] = acc;
  }
  float mx = logits[0];
#pragma unroll
  for (int v = 1; v < kV; ++v) mx = fmaxf(mx, logits[v]);
  float s = 0.0f;
#pragma unroll
  for (int v = 0; v < kV; ++v) s += expf(logits[v] - mx);
  float lse = mx + logf(s);

  score[b] += logits[tgt[b]] - lse;
}

static inline int rf_gemm_blocks(int rows, int outCols) {
  int tiles = (rows >> 4) * ((outCols + 15) >> 4);
  return (tiles + 7) / 8;                     // 8 waves (256 threads) per block
}

extern "C" void kernel_launch(void* const* d_in, const int* in_sizes, int n_in,
                              void* d_out, int out_size, void* d_ws, size_t ws_size,
                              hipStream_t stream) {
  (void)in_sizes; (void)n_in; (void)out_size; (void)ws_size;

  const int*   inputs  = (const int*)d_in[0];    // [J, T, B]
  const int*   target  = (const int*)d_in[1];    // [T, B]
  const float* encWih  = (const float*)d_in[2];  // [300, 11]
  const float* encWhh  = (const float*)d_in[3];  // [300, 100]
  const float* encBih  = (const float*)d_in[4];
  const float* encBhh  = (const float*)d_in[5];
  const float* decWih  = (const float*)d_in[6];
  const float* decWhh  = (const float*)d_in[7];
  const float* decBih  = (const float*)d_in[8];
  const float* decBhh  = (const float*)d_in[9];
  const float* Ww      = (const float*)d_in[10]; // [100, 200]
  const float* Wb      = (const float*)d_in[11];
  const float* Vw      = (const float*)d_in[12]; // [11, 100]
  const float* Vb      = (const float*)d_in[13];
  const float* Aw      = (const float*)d_in[14]; // [100, 100]
  float* score = (float*)d_out;                  // [B]

  // Workspace layout (floats). SCR (NB x 300) is time-shared by gh / AP / fc.
  float* ws   = (float*)d_ws;
  float* HBUF = ws;                                  // [T, NB, H]   = 32.77M f
  float* Hcur = HBUF + (size_t)kT * kNB * kH;        // [NB, H]
  float* Pbuf = Hcur + (size_t)kNB * kH;             // [NB, H]
  float* SCR  = Pbuf + (size_t)kNB * kH;             // [NB, 300]
  float* PAUG = SCR  + (size_t)kNB * kG3;            // [NB, 200] = [P | c]

  const int elemBlocks = (kNB * kH + 255) / 256;     // 12800

  // ---- init: h0 = 0, score = 0 (d_out/d_ws are poisoned) ----
  rf_zero<<<elemBlocks, 256, 0, stream>>>(Hcur, kNB * kH);
  rf_zero<<<(kB + 255) / 256, 256, 0, stream>>>(score, kB);

  // ---- encoder: 10 GRU steps over fused batch NB ----
  for (int t = 0; t < kT; ++t) {
    rf_gemm_wmma<<<rf_gemm_blocks(kNB, kG3), 256, 0, stream>>>(
        Hcur, kH, encWhh, kH, kG3, kNB, SCR, nullptr, 0);
    rf_gru_gate<<<elemBlocks, 256, 0, stream>>>(
        SCR, Hcur, kH, Hcur, kH, HBUF + (size_t)t * kNB * kH,
        encWih, encBih, encBhh, inputs + (size_t)t * kB, 1);
  }

  // ---- decoder init: P0 = GRU(SOS, H[:, -1]) ----
  const float* Hlast = HBUF + (size_t)(kT - 1) * kNB * kH;
  rf_gemm_wmma<<<rf_gemm_blocks(kNB, kG3), 256, 0, stream>>>(
      Hlast, kH, decWhh, kH, kG3, kNB, SCR, nullptr, 0);
  rf_gru_gate<<<elemBlocks, 256, 0, stream>>>(
      SCR, Hlast, kH, Pbuf, kH, nullptr, decWih, decBih, decBhh, nullptr, 0);

  // ---- decoder: 10 attention + output + GRU steps ----
  for (int t = 0; t < kT; ++t) {
    // AP = P @ A_w^T  (into SCR, used as [NB, H])
    rf_gemm_wmma<<<rf_gemm_blocks(kNB, kH), 256, 0, stream>>>(
        Pbuf, kH, Aw, kH, kH, kNB, SCR, nullptr, 0);
    // p_aug = [P | c] with log-softmax attention weights
    rf_attend<<<kNB / 256, 256, 0, stream>>>(HBUF, SCR, Pbuf, PAUG);
    // fc = tanh(p_aug @ W_w^T + W_b)  (into SCR, used as [NB, EMB])
    rf_gemm_wmma<<<rf_gemm_blocks(kNB, kEMB), 256, 0, stream>>>(
        PAUG, 2 * kH, Ww, 2 * kH, kEMB, kNB, SCR, Wb, 1);
    // score[b] += log_softmax(max_j fc @ V_w^T + V_b)[target[t,b]]
    rf_out_step<<<kB / 128, 128, 0, stream>>>(
        SCR, Vw, Vb, target + (size_t)t * kB, score);
    // gh = c @ dec_Whh^T (c lives at PAUG+100, stride 200) -> SCR as [NB, 300]
    rf_gemm_wmma<<<rf_gemm_blocks(kNB, kG3), 256, 0, stream>>>(
        PAUG + kH, 2 * kH, decWhh, kH, kG3, kNB, SCR, nullptr, 0);
    // P = GRU(onehot(target[t]), c)
    rf_gru_gate<<<elemBlocks, 256, 0, stream>>>(
        SCR, PAUG + kH, 2 * kH, Pbuf, kH, nullptr,
        decWih, decBih, decBhh, target + (size_t)t * kB, 2);
  }
}